// GCNModel_36103495090682
// MI455X (gfx1250) — compile-verified
//
#include <hip/hip_runtime.h>
#include <hip/hip_bf16.h>

// GraphSAGE encode, fused: gather+mean -> LDS, dual GEMM (W1,W2) via
// V_WMMA_F32_16X16X4_F32 (fp32 tensor op; problem is L2/HBM-gather bound,
// so fp32 WMMA keeps reference precision at zero cost).
//
// v2: 32-column batch tile per block (two WMMA N-tiles). Each A-fragment
// (W tile) load now feeds two WMMAs -> half the W L2 traffic, and each
// output row gets a contiguous 128B store instead of 64B.

typedef __attribute__((ext_vector_type(2))) float v2f;
typedef __attribute__((ext_vector_type(8))) float v8f;

#define N_TOTAL 100000
#define BATCH   20000
#define KNBR    16
#define D       256
#define TWO_D   512
#define OUT     256
#define BT      32            // batch columns per block = 2 WMMA N-tiles
#define LDS_STRIDE 516        // 512 + 4 floats pad -> conflict-free ds_load_b64

__global__ __launch_bounds__(256)
void gcn_encode_kernel(const int* __restrict__ nodes,
                       const int* __restrict__ neigh_idx,
                       const float* __restrict__ feat,
                       const float* __restrict__ W1,
                       const float* __restrict__ W2,
                       float* __restrict__ out)
{
    __shared__ float combined[BT * LDS_STRIDE];   // ~66 KB of 320 KB WGP LDS

    const int tid  = threadIdx.x;                 // 0..255, one feature column
    const int b0   = blockIdx.x * BT;             // batch tile base

    // ---------------- Stage 1: gather self + neighbor-mean into LDS ---------
    // Thread `tid` owns feature column d = tid for all 32 rows of the tile.
    for (int r = 0; r < BT; ++r) {
        const int b = b0 + r;
        const int self_id = nodes[b];                     // scalar (uniform)
        const float self_v = feat[self_id * D + tid];     // coalesced 1 KB row
        const int* nb = neigh_idx + b * KNBR;
        float acc = 0.0f;
        #pragma unroll
        for (int kk = 0; kk < KNBR; ++kk) {
            acc += feat[nb[kk] * D + tid];                // L2-resident gather
        }
        combined[r * LDS_STRIDE + tid]      = self_v;             // cols 0..255
        combined[r * LDS_STRIDE + D + tid]  = acc * (1.0f / 16.0f); // 256..511
    }
    __syncthreads();

    // ---------------- Stage 2: dual GEMM on the matrix pipe ------------------
    // 32 jobs = {W1,W2} x 16 output M-tiles; each job covers BOTH N-tiles of
    // this block's 32 columns. 8 waves x 4 jobs each.
    const int wave = tid >> 5;
    const int lane = tid & 31;
    const int lrow = lane & 15;    // A: M row / B: N col
    const int lhi  = lane >> 4;    // 0 -> K{0,1}, 1 -> K{2,3}

    // Per-lane base pointers for the WMMA fragments (ISA 16x16x4 f32 layout):
    //   A VGPR0/1 = W[o0+lrow][k + 2*lhi + {0,1}]
    //   B VGPR0/1 = combined[row][k + 2*lhi + {0,1}]
    const float* bbase0 = &combined[lrow * LDS_STRIDE + 2 * lhi];          // N-tile 0
    const float* bbase1 = &combined[(16 + lrow) * LDS_STRIDE + 2 * lhi];   // N-tile 1

    for (int j = wave * 4; j < wave * 4 + 4; ++j) {
        const float* W  = (j < 16) ? W1 : W2;
        const int    o0 = (j & 15) * 16;
        const float* abase = W + (o0 + lrow) * TWO_D + 2 * lhi;

        v8f acc0a = {};   // N-tile 0, even k-step
        v8f acc0b = {};   // N-tile 0, odd  k-step
        v8f acc1a = {};   // N-tile 1, even k-step
        v8f acc1b = {};   // N-tile 1, odd  k-step
        #pragma unroll 4
        for (int k = 0; k < TWO_D; k += 8) {
            v2f a0 = *(const v2f*)(abase + k);
            v2f a1 = *(const v2f*)(abase + k + 4);
            v2f b00 = *(const v2f*)(bbase0 + k);          // ds_load_b64
            v2f b01 = *(const v2f*)(bbase0 + k + 4);
            v2f b10 = *(const v2f*)(bbase1 + k);
            v2f b11 = *(const v2f*)(bbase1 + k + 4);
            acc0a = __builtin_amdgcn_wmma_f32_16x16x4_f32(
                        false, a0, false, b00, (short)0, acc0a, false, false);
            acc1a = __builtin_amdgcn_wmma_f32_16x16x4_f32(
                        false, a0, false, b10, (short)0, acc1a, false, false);
            acc0b = __builtin_amdgcn_wmma_f32_16x16x4_f32(
                        false, a1, false, b01, (short)0, acc0b, false, false);
            acc1b = __builtin_amdgcn_wmma_f32_16x16x4_f32(
                        false, a1, false, b11, (short)0, acc1b, false, false);
        }

        // ReLU + store. C/D layout: lane<16 -> (M=v,   N=lane),
        //                           lane>=16 -> (M=v+8, N=lane-16)
        float* ob = out + (size_t)((j < 16) ? 0 : 1) * OUT * BATCH;
        const int Ncol0 = b0 + lrow;        // N-tile 0 column
        const int Ncol1 = b0 + 16 + lrow;   // N-tile 1 column
        const int Mbase = o0 + 8 * lhi;
        #pragma unroll
        for (int v = 0; v < 8; ++v) {
            const size_t rowoff = (size_t)(Mbase + v) * BATCH;
            float x0 = fmaxf(acc0a[v] + acc0b[v], 0.0f);
            float x1 = fmaxf(acc1a[v] + acc1b[v], 0.0f);
            ob[rowoff + Ncol0] = x0;
            ob[rowoff + Ncol1] = x1;
        }
    }
}

extern "C" void kernel_launch(void* const* d_in, const int* in_sizes, int n_in,
                              void* d_out, int out_size, void* d_ws, size_t ws_size,
                              hipStream_t stream) {
    const int*   nodes = (const int*)  d_in[0];   // [20000]
    const int*   neigh = (const int*)  d_in[1];   // [20000,16]
    const float* feat  = (const float*)d_in[2];   // [100000,256]
    const float* W1    = (const float*)d_in[3];   // [256,512]
    const float* W2    = (const float*)d_in[4];   // [256,512]
    float*       out   = (float*)d_out;           // [2,256,20000]

    dim3 grid(BATCH / BT);   // 625 blocks
    dim3 block(256);         // 8 wave32s
    gcn_encode_kernel<<<grid, block, 0, stream>>>(nodes, neigh, feat, W1, W2, out);
}